// CombinedLoss_17635135717907
// MI455X (gfx1250) — compile-verified
//
#include <hip/hip_runtime.h>
#include <math.h>

// Problem constants (reference: B=16, C=8, H=512, W=512)
#define Bn   16
#define Cn   8
#define Hn   512
#define Wn   512
#define TILE 16
#define LW   18              // tile + 1-halo on each side
#define PST  9               // LDS prob stride (pad to break bank conflicts)
#define NBLK (Bn * (Hn / TILE) * (Wn / TILE))   // 16384 blocks

#define RWAVES 8             // waves in the final-reduce block
#define CHUNK  (NBLK / RWAVES)        // 2048 partials per wave per quantity
#define ITERS  (CHUNK / 64)           // 32 wmma iterations per wave/quantity

typedef __attribute__((ext_vector_type(2))) float v2f;
typedef __attribute__((ext_vector_type(8))) float v8f;

// ---------------------------------------------------------------------------
// Pass 1: one block per 16x16 tile. Cooperatively build softmax probs, argmax
// class and NLL for the 18x18 haloed region in LDS, then accumulate the three
// loss partial sums and write per-block partials (deterministic, no atomics).
// ---------------------------------------------------------------------------
__global__ __launch_bounds__(256) void fused_loss_partials(
    const float* __restrict__ logits,   // [B, C, H, W]
    const int*   __restrict__ labels,   // [B, H, W]
    float*       __restrict__ parts)    // [3 * NBLK]
{
    __shared__ float sp[LW * LW * PST];   // probs (8 used of 9 per pixel)
    __shared__ float snll[LW * LW];       // -log_softmax at label
    __shared__ int   scls[LW * LW];       // argmax class, -1 if out of image
    __shared__ float sred[3 * 256];

    const int blk = blockIdx.x;
    const int tw  = blk & 31;            // W/16 = 32 tiles
    const int th  = (blk >> 5) & 31;     // H/16 = 32 tiles
    const int b   = blk >> 10;
    const int h0  = th * TILE, w0 = tw * TILE;
    const int tid = threadIdx.x;

    const size_t plane = (size_t)Hn * Wn;
    const float* base  = logits + (size_t)b * Cn * plane;
    const int*   lbase = labels + (size_t)b * plane;

    // ---- cooperative load + per-pixel softmax for 18x18 region ----
    for (int idx = tid; idx < LW * LW; idx += 256) {
        const int lr = idx / LW, lc = idx - lr * LW;
        const int gh = h0 + lr - 1, gw = w0 + lc - 1;
        int   cls = -1;
        float nll = 0.0f;
        float p[Cn];
        #pragma unroll
        for (int c = 0; c < Cn; ++c) p[c] = 0.0f;

        if (gh >= 0 && gh < Hn && gw >= 0 && gw < Wn) {
            const size_t off = (size_t)gh * Wn + gw;
            float x[Cn];
            #pragma unroll
            for (int c = 0; c < Cn; ++c) x[c] = base[(size_t)c * plane + off];
            float m = x[0];
            cls = 0;
            #pragma unroll
            for (int c = 1; c < Cn; ++c)
                if (x[c] > m) { m = x[c]; cls = c; }   // first-max == jnp.argmax
            float s = 0.0f;
            #pragma unroll
            for (int c = 0; c < Cn; ++c) { p[c] = __expf(x[c] - m); s += p[c]; }
            const float inv = 1.0f / s;
            #pragma unroll
            for (int c = 0; c < Cn; ++c) p[c] *= inv;
            const int lab = lbase[off] & 7;
            nll = m + logf(s) - x[lab];                // -log_softmax[label]
        }
        scls[idx] = cls;
        snll[idx] = nll;
        #pragma unroll
        for (int c = 0; c < Cn; ++c) sp[idx * PST + c] = p[c];
    }
    __syncthreads();

    // ---- per-thread interior pixel contributions ----
    const int ti = tid >> 4, tj = tid & 15;
    const int li = ti + 1,  lj = tj + 1;
    const int gh = h0 + ti, gw = w0 + tj;              // always in-bounds
    const int pix = li * LW + lj;

    float ce   = snll[pix];
    float edge = 0.0f;
    float cons = 0.0f;

    if (gh + 1 < Hn) {                                  // diff along H (down)
        #pragma unroll
        for (int c = 0; c < Cn; ++c)
            edge += fabsf(sp[pix * PST + c] - sp[(pix + LW) * PST + c]);
    }
    if (gw + 1 < Wn) {                                  // diff along W (right)
        #pragma unroll
        for (int c = 0; c < Cn; ++c)
            edge += fabsf(sp[pix * PST + c] - sp[(pix + 1) * PST + c]);
    }

    {
        const int pc = scls[pix];
        const int up = scls[pix - LW], dn = scls[pix + LW];
        const int lf = scls[pix - 1],  rt = scls[pix + 1];
        const bool upv = gh > 0, dnv = gh < Hn - 1;
        const bool lfv = gw > 0, rtv = gw < Wn - 1;
        const int ref = upv ? up : dn;                  // H>=2: one always valid
        const bool alleq = ((up == ref) || !upv) && ((dn == ref) || !dnv) &&
                           ((lf == ref) || !lfv) && ((rt == ref) || !rtv);
        if (alleq && (pc != ref)) cons = 1.0f;
    }

    // ---- deterministic block tree-reduction ----
    sred[tid]       = ce;
    sred[256 + tid] = edge;
    sred[512 + tid] = cons;
    __syncthreads();
    for (int s = 128; s > 0; s >>= 1) {
        if (tid < s) {
            sred[tid]       += sred[tid + s];
            sred[256 + tid] += sred[256 + tid + s];
            sred[512 + tid] += sred[512 + tid + s];
        }
        __syncthreads();
    }
    if (tid == 0) {
        parts[blk]            = sred[0];
        parts[NBLK + blk]     = sred[256];
        parts[2 * NBLK + blk] = sred[512];
    }
}

// ---------------------------------------------------------------------------
// Pass 2: one 256-thread block (8 wave32s). Each wave reduces a fixed
// contiguous chunk of 2048 partials per quantity with V_WMMA_F32_16X16X4_F32:
// D = ones(16x4) x B(4x16 of data) + C. Every B element contributes exactly
// once to the column sums; summing D's VGPR0 over all 32 lanes counts each
// column sum twice -> wave total = 0.5 * waveSum(acc[0]). Loads are software-
// pipelined (preload next b64 before the wmma) so s_wait overlaps the matrix
// op, and the 8 waves interleave on the SIMDs to hide latency. Wave totals
// are combined by thread 0 in a fixed order -> bit-deterministic.
// ---------------------------------------------------------------------------
__global__ __launch_bounds__(256) void final_reduce(
    const float* __restrict__ parts, float* __restrict__ out)
{
    __shared__ float sl[3 * RWAVES];

    const int tid  = threadIdx.x;
    const int wv   = tid >> 5;
    const int lane = tid & 31;
    const v2f ones = {1.0f, 1.0f};

    #pragma unroll 1
    for (int q = 0; q < 3; ++q) {
        v8f acc = {};
        const float* p = parts + (size_t)q * NBLK + (size_t)wv * CHUNK + 2 * lane;

        v2f bv = *(const v2f*)(p);                       // prologue load
        #pragma unroll 1
        for (int it = 0; it < ITERS - 1; ++it) {
            v2f nxt = *(const v2f*)(p + (it + 1) * 64);  // prefetch next
            acc = __builtin_amdgcn_wmma_f32_16x16x4_f32(
                false, ones, false, bv, (short)0, acc, false, false);
            bv = nxt;
        }
        acc = __builtin_amdgcn_wmma_f32_16x16x4_f32(
            false, ones, false, bv, (short)0, acc, false, false);

        float v = acc[0];
        #pragma unroll
        for (int off = 16; off > 0; off >>= 1) v += __shfl_xor(v, off);
        if (lane == 0) sl[q * RWAVES + wv] = 0.5f * v;
    }
    __syncthreads();

    if (tid == 0) {
        float tot[3];
        #pragma unroll
        for (int q = 0; q < 3; ++q) {
            float t = 0.0f;
            #pragma unroll
            for (int w = 0; w < RWAVES; ++w) t += sl[q * RWAVES + w];
            tot[q] = t;
        }
        const float inv_bhw = 1.0f / (float)((long long)Bn * Hn * Wn);
        const float inv_hw  = 1.0f / (float)(Hn * Wn);
        out[0] = 1.1f   * (tot[0] * inv_bhw)    // W_CE  * ce mean
               + 0.001f * (tot[1] * inv_hw)     // W_EDGE* edge / (H*W)
               + 0.1f   * (tot[2] * inv_bhw);   // W_CONS* cons / (B*H*W)
    }
}

extern "C" void kernel_launch(void* const* d_in, const int* in_sizes, int n_in,
                              void* d_out, int out_size, void* d_ws, size_t ws_size,
                              hipStream_t stream) {
    const float* logits = (const float*)d_in[0];
    const int*   labels = (const int*)d_in[1];
    float*       out    = (float*)d_out;
    float*       parts  = (float*)d_ws;       // needs 3*16384*4 = 192 KiB

    fused_loss_partials<<<NBLK, 256, 0, stream>>>(logits, labels, parts);
    final_reduce<<<1, 256, 0, stream>>>(parts, out);
}